// SuperLoss_43035572306155
// MI455X (gfx1250) — compile-verified
//
#include <hip/hip_runtime.h>
#include <math.h>

typedef __attribute__((ext_vector_type(2))) float v2f;
typedef __attribute__((ext_vector_type(8))) float v8f;

#define HAS_WMMA_F32_16X16X4 __has_builtin(__builtin_amdgcn_wmma_f32_16x16x4_f32)

// ---------------------------------------------------------------------------
// Kernel 1: per-block partial sums of `loss`, reduction driven through
// V_WMMA_F32_16X16X4_F32 (D = A * ones + C). Each WMMA folds 64 fp32 values
// (2 per lane) into the 16x16 accumulator; since B == 1, every accumulator
// column holds the same row-sum, so total = (sum of all cells) / 16 exactly.
// 512 blocks x 256 threads (8 waves). Each wave consumes 4096 elements
// (64 WMMA iterations x 64 floats), fully coalesced (256B per wave per load).
// ---------------------------------------------------------------------------
__global__ __launch_bounds__(256) void sl_partial_sum_kernel(
    const float* __restrict__ loss, float* __restrict__ partials,
    int chunk_per_block)
{
    const int wave = threadIdx.x >> 5;
    const int lane = threadIdx.x & 31;
    const int wave_chunk = chunk_per_block >> 3;     // 8 waves per block
    const int iters = wave_chunk >> 6;               // 64 floats per WMMA
    const long base = (long)blockIdx.x * chunk_per_block + (long)wave * wave_chunk;

    float laneSum;

#if HAS_WMMA_F32_16X16X4
    v8f acc = {};
    v2f ones; ones.x = 1.0f; ones.y = 1.0f;
    const float* p = loss + base + lane * 2;
    for (int it = 0; it < iters; ++it) {
        v2f a = *(const v2f*)(p + (long)it * 64);
        // 8 args: (neg_a, A, neg_b, B, c_mod, C, reuse_a, reuse_b)
        acc = __builtin_amdgcn_wmma_f32_16x16x4_f32(
            /*neg_a=*/false, a, /*neg_b=*/false, ones,
            /*c_mod=*/(short)0, acc, /*reuse_a=*/false, /*reuse_b=*/false);
    }
    laneSum = ((acc[0] + acc[1]) + (acc[2] + acc[3])) +
              ((acc[4] + acc[5]) + (acc[6] + acc[7]));
#else
    float s = 0.0f;
    const float* p = loss + base + lane * 2;
    for (int it = 0; it < iters; ++it) {
        v2f a = *(const v2f*)(p + (long)it * 64);
        s += a.x + a.y;
    }
    laneSum = s;
#endif

    // wave32 cross-lane tree reduce (deterministic fixed order)
    #pragma unroll
    for (int off = 16; off > 0; off >>= 1)
        laneSum += __shfl_xor(laneSum, off, 32);

#if HAS_WMMA_F32_16X16X4
    laneSum *= (1.0f / 16.0f);   // each column duplicated 16x; exact scale
#endif

    __shared__ float smem[8];
    if (lane == 0) smem[wave] = laneSum;
    __syncthreads();
    if (threadIdx.x == 0) {
        float s = 0.0f;
        #pragma unroll
        for (int w = 0; w < 8; ++w) s += smem[w];
        partials[blockIdx.x] = s;
    }
}

// ---------------------------------------------------------------------------
// Kernel 2: single block reduces the 512 block partials and computes
// tau = (1 - MOM) * TAU0 + MOM * mean   (MOM = 0.1, TAU0 = 0.5)
// ---------------------------------------------------------------------------
__global__ __launch_bounds__(512) void sl_finalize_tau_kernel(
    const float* __restrict__ partials, float* __restrict__ tau_out,
    int nparts, float inv_n)
{
    float v = (threadIdx.x < (unsigned)nparts) ? partials[threadIdx.x] : 0.0f;
    #pragma unroll
    for (int off = 16; off > 0; off >>= 1)
        v += __shfl_xor(v, off, 32);

    __shared__ float smem[16];
    const int wave = threadIdx.x >> 5;
    const int lane = threadIdx.x & 31;
    if (lane == 0) smem[wave] = v;
    __syncthreads();
    if (threadIdx.x == 0) {
        float s = 0.0f;
        #pragma unroll
        for (int w = 0; w < 16; ++w) s += smem[w];
        const float mean = s * inv_n;
        tau_out[0] = 0.9f * 0.5f + 0.1f * mean;
    }
}

// ---------------------------------------------------------------------------
// Kernel 3: elementwise SuperLoss. float4 streaming; Lambert-W via 4 Halley
// iterations (cubic convergence from the reference's initializer reaches fp32
// precision in <=3 steps). Fast-path transcendentals: __expf, __fdividef.
// ---------------------------------------------------------------------------
__device__ __forceinline__ float sl_lambertw(float z)
{
    // init: branch-point series near z ~ -1/e, log1p elsewhere
    const float p2 = fmaxf(fmaf(5.43656365691809f, z, 2.0f), 0.0f); // 2*(e*z+1)
    const float p  = __builtin_sqrtf(p2);
    const float w_near = -1.0f + p * (1.0f - p * (1.0f / 3.0f));
    float w = (z < -0.25f) ? w_near : log1pf(z);
    #pragma unroll
    for (int it = 0; it < 4; ++it) {
        const float ew   = __expf(w);
        const float f    = fmaf(w, ew, -z);
        const float wp1  = w + 1.0f;
        const float corr = __fdividef((w + 2.0f) * f, 2.0f * wp1);
        const float den  = fmaf(ew, wp1, -corr);
        w = w - __fdividef(f, den);
    }
    return w;
}

__global__ __launch_bounds__(256) void sl_superloss_kernel(
    const float* __restrict__ loss, const float* __restrict__ tau_ptr,
    float* __restrict__ out_super, float* __restrict__ out_sigma, int n4)
{
    const int i = blockIdx.x * blockDim.x + threadIdx.x;
    if (i >= n4) return;

    const float tau = tau_ptr[0];
    // z_min = -exp(-1) + FLT_EPSILON (constant-folded)
    const float zmin = -0.36787944117144233f + 1.1920928955078125e-7f;

    const float4 x = ((const float4*)loss)[i];
    float xs[4] = {x.x, x.y, x.z, x.w};
    float sg[4], sl[4];

    #pragma unroll
    for (int k = 0; k < 4; ++k) {
        const float beta = xs[k] - tau;            // LAM = 1
        const float z = fmaxf(zmin, 0.5f * beta);
        const float w = sl_lambertw(z);
        const float sigma = __expf(-w);
        sg[k] = sigma;
        sl[k] = sigma * xs[k];
    }

    float4 o;
    o.x = sl[0]; o.y = sl[1]; o.z = sl[2]; o.w = sl[3];
    ((float4*)out_super)[i] = o;
    o.x = sg[0]; o.y = sg[1]; o.z = sg[2]; o.w = sg[3];
    ((float4*)out_sigma)[i] = o;
}

// ---------------------------------------------------------------------------
extern "C" void kernel_launch(void* const* d_in, const int* in_sizes, int n_in,
                              void* d_out, int out_size, void* d_ws, size_t ws_size,
                              hipStream_t stream)
{
    const float* loss = (const float*)d_in[0];
    const int n = in_sizes[0];                 // 16,777,216 (2^24)

    float* out_super = (float*)d_out;          // [0, n)
    float* out_sigma = out_super + n;          // [n, 2n)

    float* partials = (float*)d_ws;            // 512 floats
    float* tau_ptr  = partials + 512;          // 1 float

    const int RBLOCKS = 512;
    const int chunk = n / RBLOCKS;             // 32768 (mult of 8 waves * 64)

    sl_partial_sum_kernel<<<RBLOCKS, 256, 0, stream>>>(loss, partials, chunk);
    sl_finalize_tau_kernel<<<1, 512, 0, stream>>>(partials, tau_ptr, RBLOCKS,
                                                  1.0f / (float)n);

    const int n4 = n >> 2;
    sl_superloss_kernel<<<(n4 + 255) / 256, 256, 0, stream>>>(
        loss, tau_ptr, out_super, out_sigma, n4);
}